// GRN_71743133712529
// MI455X (gfx1250) — compile-verified
//
#include <hip/hip_runtime.h>
#include <hip/hip_bf16.h>
#include <math.h>

#define B_DIM 64
#define N_NODES 512
#define D_DIM 64
#define TOPK 20
#define NEG_SLOPE 0.2f
#define EPS 1e-5f
#define NROWS (B_DIM * N_NODES)   // 32768

typedef __attribute__((ext_vector_type(2))) float v2f;
typedef __attribute__((ext_vector_type(8))) float v8f;

// ---------------------------------------------------------------------------
// Kernel 1: cosine top-k over emb_table + per-node att_em dot products
// ---------------------------------------------------------------------------
__global__ void topk_kernel(const float* __restrict__ w,
                            const float* __restrict__ att_em_i,
                            const float* __restrict__ att_em_j,
                            int* __restrict__ topk_idx,
                            float* __restrict__ s_em_i,
                            float* __restrict__ s_em_j) {
  __shared__ float wi[D_DIM];
  __shared__ float scores[N_NODES];
  __shared__ float rv[256];
  __shared__ int   ri[256];
  const int i = blockIdx.x;
  const int t = threadIdx.x;
  if (t < D_DIM) wi[t] = w[i * D_DIM + t];
  __syncthreads();

  float ni = 0.f;
  #pragma unroll
  for (int d = 0; d < D_DIM; ++d) { float v = wi[d]; ni += v * v; }

  for (int j = t; j < N_NODES; j += 256) {
    float dot = 0.f, nj = 0.f;
    for (int d = 0; d < D_DIM; ++d) {
      float vj = w[j * D_DIM + d];
      dot += wi[d] * vj;
      nj  += vj * vj;
    }
    scores[j] = dot * rsqrtf(ni * nj);
  }
  if (t == 0) {
    float si = 0.f, sj = 0.f;
    for (int d = 0; d < D_DIM; ++d) { si += wi[d] * att_em_i[d]; sj += wi[d] * att_em_j[d]; }
    s_em_i[i] = si;
    s_em_j[i] = sj;
  }
  __syncthreads();

  for (int k = 0; k < TOPK; ++k) {
    float v0 = scores[t], v1 = scores[t + 256];
    float bv; int bi;
    if (v1 > v0) { bv = v1; bi = t + 256; } else { bv = v0; bi = t; }
    rv[t] = bv; ri[t] = bi;
    __syncthreads();
    for (int s = 128; s > 0; s >>= 1) {
      if (t < s) {
        float ov = rv[t + s]; int oi = ri[t + s];
        if (ov > rv[t] || (ov == rv[t] && oi < ri[t])) { rv[t] = ov; ri[t] = oi; }
      }
      __syncthreads();
    }
    if (t == 0) {
      topk_idx[i * TOPK + k] = ri[0];
      scores[ri[0]] = -1e30f;
    }
    __syncthreads();
  }
}

// ---------------------------------------------------------------------------
// Kernel 2: xw = data @ W_gnn^T   (32768x64) x (64x64), WMMA fp32 16x16x4
// ---------------------------------------------------------------------------
__global__ void xw_gemm_kernel(const float* __restrict__ A,   // data rows (NROWS x 64)
                               const float* __restrict__ Wg,  // W_gnn (64 x 64) row-major
                               float* __restrict__ xw) {
  __shared__ float As[16][65];
  const int tid  = threadIdx.x;
  const int lane = tid & 31;
  const int wv   = tid >> 5;                 // 0..3 -> output column tile
  const int rowBase = blockIdx.x * 16;

  for (int idx = tid; idx < 16 * 64; idx += 128) {
    As[idx >> 6][idx & 63] = A[(size_t)(rowBase + (idx >> 6)) * 64 + (idx & 63)];
  }
  __syncthreads();

  const int half = lane >> 4;
  const int mn   = lane & 15;
  const int colBase = wv * 16;

  v8f c = {};
  #pragma unroll
  for (int k0 = 0; k0 < 64; k0 += 4) {
    const int kA = k0 + 2 * half;
    v2f a, b;
    a.x = As[mn][kA];
    a.y = As[mn][kA + 1];
    // B[k][n] = W_gnn[n][k]
    b.x = Wg[(colBase + mn) * 64 + kA];
    b.y = Wg[(colBase + mn) * 64 + kA + 1];
    c = __builtin_amdgcn_wmma_f32_16x16x4_f32(false, a, false, b, (short)0, c, false, false);
  }
  #pragma unroll
  for (int r = 0; r < 8; ++r) {
    const int mr = r + 8 * half;
    xw[(size_t)(rowBase + mr) * 64 + colBase + mn] = c[r];
  }
}

// ---------------------------------------------------------------------------
// Kernel 3: attention + neighbor aggregation, one wave32 per (b,n) row.
// Also accumulates BN1 channel sums (sum, sumsq) via LDS + global atomics.
// ---------------------------------------------------------------------------
__global__ void attn_kernel(const float* __restrict__ xw,
                            const int* __restrict__ topk_idx,
                            const float* __restrict__ att_i,
                            const float* __restrict__ att_j,
                            const float* __restrict__ s_em_i,
                            const float* __restrict__ s_em_j,
                            const float* __restrict__ bias_gnn,
                            float* __restrict__ out1,
                            float* __restrict__ stat_sum1,
                            float* __restrict__ stat_sq1) {
  __shared__ float attjS[64];
  __shared__ float ssum[64], ssq[64];
  const int tid  = threadIdx.x;
  const int lane = tid & 31;
  const int wv   = tid >> 5;
  if (tid < 64) { attjS[tid] = att_j[tid]; ssum[tid] = 0.f; ssq[tid] = 0.f; }
  __syncthreads();

  const int row   = blockIdx.x * 4 + wv;    // 0..32767
  const int n     = row & (N_NODES - 1);
  const int bBase = row & ~(N_NODES - 1);   // b * N
  const float* xr = xw + (size_t)row * 64;
  float x0 = xr[lane], x1 = xr[lane + 32];

  // a_i = xw[row] . att_i + s_em_i[n]
  float p = x0 * att_i[lane] + x1 * att_i[lane + 32];
  #pragma unroll
  for (int off = 16; off; off >>= 1) p += __shfl_xor(p, off, 32);
  const float ai = p + s_em_i[n];

  // lane k < 20 computes its neighbor score
  float e = -1e30f;
  int myJ = 0;
  if (lane < TOPK) {
    myJ = topk_idx[n * TOPK + lane];
    const float* xj = xw + (size_t)(bBase + myJ) * 64;
    float aj = 0.f;
    for (int d = 0; d < 64; ++d) aj += xj[d] * attjS[d];
    aj += s_em_j[myJ];
    float t = ai + aj;
    e = (t > 0.f) ? t : NEG_SLOPE * t;
  }
  // softmax across 20 lanes
  float mx = e;
  #pragma unroll
  for (int off = 16; off; off >>= 1) mx = fmaxf(mx, __shfl_xor(mx, off, 32));
  float ex = (lane < TOPK) ? __expf(e - mx) : 0.f;
  float sm = ex;
  #pragma unroll
  for (int off = 16; off; off >>= 1) sm += __shfl_xor(sm, off, 32);
  const float alpha = ex / sm;

  // weighted gather: each lane owns channels (lane, lane+32)
  float acc0 = 0.f, acc1 = 0.f;
  for (int k = 0; k < TOPK; ++k) {
    float ak = __shfl(alpha, k, 32);
    int   jk = __shfl(myJ, k, 32);
    const float* xj = xw + (size_t)(bBase + jk) * 64;
    acc0 += ak * xj[lane];
    acc1 += ak * xj[lane + 32];
  }
  acc0 += bias_gnn[lane];
  acc1 += bias_gnn[lane + 32];
  out1[(size_t)row * 64 + lane]      = acc0;
  out1[(size_t)row * 64 + lane + 32] = acc1;

  atomicAdd(&ssum[lane], acc0);       atomicAdd(&ssq[lane], acc0 * acc0);
  atomicAdd(&ssum[lane + 32], acc1);  atomicAdd(&ssq[lane + 32], acc1 * acc1);
  __syncthreads();
  if (tid < 64) {
    atomicAdd(&stat_sum1[tid], ssum[tid]);
    atomicAdd(&stat_sq1[tid],  ssq[tid]);
  }
}

// ---------------------------------------------------------------------------
// Kernel 4: BN1 -> relu -> * w ; accumulate BN2 stats
// ---------------------------------------------------------------------------
__global__ void post_kernel(const float* __restrict__ out1,
                            const float* __restrict__ w,
                            const float* __restrict__ bn1_g, const float* __restrict__ bn1_b,
                            const float* __restrict__ stat_sum1, const float* __restrict__ stat_sq1,
                            float* __restrict__ out2,
                            float* __restrict__ stat_sum2, float* __restrict__ stat_sq2) {
  __shared__ float ssum[64], ssq[64];
  const int tid = threadIdx.x;
  if (tid < 64) { ssum[tid] = 0.f; ssq[tid] = 0.f; }
  __syncthreads();
  const float invM = 1.f / (float)NROWS;
  const size_t total = (size_t)NROWS * 64;
  for (size_t idx = (size_t)blockIdx.x * 256 + tid; idx < total; idx += (size_t)gridDim.x * 256) {
    const int d = (int)(idx & 63);
    const int n = (int)((idx >> 6) & (N_NODES - 1));
    float mean = stat_sum1[d] * invM;
    float var  = stat_sq1[d] * invM - mean * mean;
    float v = (out1[idx] - mean) * rsqrtf(var + EPS) * bn1_g[d] + bn1_b[d];
    v = fmaxf(v, 0.f);
    v *= w[n * 64 + d];
    out2[idx] = v;
    atomicAdd(&ssum[d], v);
    atomicAdd(&ssq[d], v * v);
  }
  __syncthreads();
  if (tid < 64) {
    atomicAdd(&stat_sum2[tid], ssum[tid]);
    atomicAdd(&stat_sq2[tid],  ssq[tid]);
  }
}

// ---------------------------------------------------------------------------
// Kernel 5: BN2 -> relu -> lin (64->32) -> out (32->64), WMMA fp32, 1 wave/16 rows
// ---------------------------------------------------------------------------
__global__ void head_kernel(const float* __restrict__ out2,
                            const float* __restrict__ bn2_g, const float* __restrict__ bn2_b,
                            const float* __restrict__ stat_sum2, const float* __restrict__ stat_sq2,
                            const float* __restrict__ linW, const float* __restrict__ linB,
                            const float* __restrict__ outW, const float* __restrict__ outB,
                            float* __restrict__ y) {
  __shared__ float act[16][65];
  __shared__ float hs[16][33];
  const int lane = threadIdx.x;            // 32 threads
  const int rowBase = blockIdx.x * 16;
  const float invM = 1.f / (float)NROWS;

  for (int idx = lane; idx < 16 * 64; idx += 32) {
    const int r = idx >> 6, d = idx & 63;
    float mean = stat_sum2[d] * invM;
    float var  = stat_sq2[d] * invM - mean * mean;
    float v = (out2[(size_t)(rowBase + r) * 64 + d] - mean) * rsqrtf(var + EPS) * bn2_g[d] + bn2_b[d];
    act[r][d] = fmaxf(v, 0.f);
  }
  __syncthreads();

  const int half = lane >> 4;
  const int mn   = lane & 15;

  // GEMM1: act(16x64) @ lin_W^T(64x32) -> hs(16x32), + lin_b
  #pragma unroll
  for (int ct = 0; ct < 2; ++ct) {
    v8f c = {};
    #pragma unroll
    for (int k0 = 0; k0 < 64; k0 += 4) {
      const int kA = k0 + 2 * half;
      v2f a, b;
      a.x = act[mn][kA];
      a.y = act[mn][kA + 1];
      b.x = linW[(ct * 16 + mn) * 64 + kA];
      b.y = linW[(ct * 16 + mn) * 64 + kA + 1];
      c = __builtin_amdgcn_wmma_f32_16x16x4_f32(false, a, false, b, (short)0, c, false, false);
    }
    #pragma unroll
    for (int r = 0; r < 8; ++r) {
      const int mr = r + 8 * half;
      hs[mr][ct * 16 + mn] = c[r] + linB[ct * 16 + mn];
    }
  }
  __syncthreads();

  // GEMM2: hs(16x32) @ out_W^T(32x64) -> y, + out_b
  #pragma unroll
  for (int ct = 0; ct < 4; ++ct) {
    v8f c = {};
    #pragma unroll
    for (int k0 = 0; k0 < 32; k0 += 4) {
      const int kA = k0 + 2 * half;
      v2f a, b;
      a.x = hs[mn][kA];
      a.y = hs[mn][kA + 1];
      b.x = outW[(ct * 16 + mn) * 32 + kA];
      b.y = outW[(ct * 16 + mn) * 32 + kA + 1];
      c = __builtin_amdgcn_wmma_f32_16x16x4_f32(false, a, false, b, (short)0, c, false, false);
    }
    #pragma unroll
    for (int r = 0; r < 8; ++r) {
      const int mr = r + 8 * half;
      y[(size_t)(rowBase + mr) * 64 + ct * 16 + mn] = c[r] + outB[ct * 16 + mn];
    }
  }
}

// ---------------------------------------------------------------------------
extern "C" void kernel_launch(void* const* d_in, const int* in_sizes, int n_in,
                              void* d_out, int out_size, void* d_ws, size_t ws_size,
                              hipStream_t stream) {
  const float* data     = (const float*)d_in[0];
  // d_in[1] = org_edge_index (unused by the reference)
  const float* emb      = (const float*)d_in[2];
  const float* Wg       = (const float*)d_in[3];
  const float* att_i    = (const float*)d_in[4];
  const float* att_j    = (const float*)d_in[5];
  const float* att_em_i = (const float*)d_in[6];
  const float* att_em_j = (const float*)d_in[7];
  const float* bias     = (const float*)d_in[8];
  const float* bn1_g    = (const float*)d_in[9];
  const float* bn1_b    = (const float*)d_in[10];
  const float* bn2_g    = (const float*)d_in[11];
  const float* bn2_b    = (const float*)d_in[12];
  const float* linW     = (const float*)d_in[13];
  const float* linB     = (const float*)d_in[14];
  const float* outW     = (const float*)d_in[15];
  const float* outB     = (const float*)d_in[16];
  float* y = (float*)d_out;

  char* ws = (char*)d_ws;
  int*   topk   = (int*)ws;                       // 512*20*4 = 40960 B
  float* s_em_i = (float*)(ws + 40960);           // 512 floats
  float* s_em_j = (float*)(ws + 43008);           // 512 floats
  float* stats  = (float*)(ws + 45056);           // 256 floats (sum1|sq1|sum2|sq2)
  float* sum1 = stats;
  float* sq1  = stats + 64;
  float* sum2 = stats + 128;
  float* sq2  = stats + 192;
  float* xw   = (float*)(ws + 49152);             // 8 MiB
  float* out1 = xw   + (size_t)NROWS * 64;        // 8 MiB
  float* out2 = out1 + (size_t)NROWS * 64;        // 8 MiB

  hipMemsetAsync(stats, 0, 256 * sizeof(float), stream);
  topk_kernel   <<<N_NODES, 256, 0, stream>>>(emb, att_em_i, att_em_j, topk, s_em_i, s_em_j);
  xw_gemm_kernel<<<NROWS / 16, 128, 0, stream>>>(data, Wg, xw);
  attn_kernel   <<<NROWS / 4, 128, 0, stream>>>(xw, topk, att_i, att_j, s_em_i, s_em_j,
                                                bias, out1, sum1, sq1);
  post_kernel   <<<2048, 256, 0, stream>>>(out1, emb, bn1_g, bn1_b, sum1, sq1,
                                           out2, sum2, sq2);
  head_kernel   <<<NROWS / 16, 32, 0, stream>>>(out2, bn2_g, bn2_b, sum2, sq2,
                                                linW, linB, outW, outB, y);
}